// LearnedRouter_89129161326933
// MI455X (gfx1250) — compile-verified
//
#include <hip/hip_runtime.h>
#include <math.h>

typedef __attribute__((ext_vector_type(2))) float v2f;
typedef __attribute__((ext_vector_type(8))) float v8f;
typedef __attribute__((ext_vector_type(4))) int   i4;

#define D_MODEL 2048
#define N_SETS  64
#define SEQ     4096
#define CHUNK   256
#define XPAD    4
#define NCHUNK  (D_MODEL / CHUNK)

#define AS1 __attribute__((address_space(1)))
#define AS3 __attribute__((address_space(3)))

#if __has_builtin(__builtin_amdgcn_global_load_async_to_lds_b128)
#define HAVE_ASYNC_LDS 1
#endif

#if __has_builtin(__builtin_amdgcn_s_wait_asynccnt)
#define WAIT_ASYNC(n) __builtin_amdgcn_s_wait_asynccnt(n)
#else
#define WAIT_ASYNC(n) asm volatile("s_wait_asynccnt " #n ::: "memory")
#endif

// ---------------------------------------------------------------------------
// Kernel A: Mt[n][d] = scale * sum_e W_q[e][d] * router[n][e]
// grid = 128 (d-tiles of 16), block = 128 (4 waves, wave w = n-tile w)
// ---------------------------------------------------------------------------
__global__ __launch_bounds__(128) void lr_precompute_M(const float* __restrict__ Wq,
                                                       const float* __restrict__ router,
                                                       float* __restrict__ Mt,
                                                       float scale) {
    const int lane = threadIdx.x & 31;
    const int w    = threadIdx.x >> 5;     // n-tile 0..3
    const int l15  = lane & 15;
    const int half = lane >> 4;
    const int kb   = half * 2;             // K pair selector per ISA A-layout
    const int dA   = blockIdx.x * 16 + l15;          // A row (d index)
    const int nB   = w * 16 + l15;                   // B col (n index)

    v8f acc = {};
    const float* brow = router + (size_t)nB * D_MODEL + kb;
#pragma unroll 4
    for (int k = 0; k < D_MODEL; k += 4) {
        v2f a, b;
        // A[d,e] = Wq[e,d]; lane needs e = k+kb, k+kb+1 (column walk, lane-coalesced)
        a.x = Wq[(size_t)(k + kb) * D_MODEL + dA];
        a.y = Wq[(size_t)(k + kb + 1) * D_MODEL + dA];
        // B[e,n] = router[n,e]; contiguous pair -> b64
        b = *(const v2f*)(brow + k);
        acc = __builtin_amdgcn_wmma_f32_16x16x4_f32(false, a, false, b,
                                                    (short)0, acc, false, false);
    }
#pragma unroll
    for (int j = 0; j < 8; ++j) {
        int d = blockIdx.x * 16 + j + 8 * half;
        Mt[(size_t)nB * D_MODEL + d] = acc[j] * scale;
    }
}

// ---------------------------------------------------------------------------
// Kernel A2: biasv[n] = scale * dot(b_q, router[n,:])   (grid=64, block=256)
// ---------------------------------------------------------------------------
__global__ __launch_bounds__(256) void lr_bias(const float* __restrict__ bq,
                                               const float* __restrict__ router,
                                               float* __restrict__ biasv,
                                               float scale) {
    __shared__ float red[256];
    const int n = blockIdx.x;
    float s = 0.f;
    for (int e = threadIdx.x; e < D_MODEL; e += 256)
        s += bq[e] * router[(size_t)n * D_MODEL + e];
    red[threadIdx.x] = s;
    __syncthreads();
    for (int st = 128; st > 0; st >>= 1) {
        if ((int)threadIdx.x < st) red[threadIdx.x] += red[threadIdx.x + st];
        __syncthreads();
    }
    if (threadIdx.x == 0) biasv[n] = red[0] * scale;
}

// ---------------------------------------------------------------------------
// Kernel B: fused scores-GEMM + mask + top2 + softmax + weighted gather.
// grid = B*S/16 = 1024 blocks of 16 tokens; block = 128 (4 waves = 4 n-tiles)
// x-tile staged via double-buffered GLOBAL_LOAD_ASYNC_TO_LDS_B128 (ASYNCcnt).
// ---------------------------------------------------------------------------
__global__ __launch_bounds__(128) void lr_router_main(const float* __restrict__ x,
                                                      const float* __restrict__ set_states,
                                                      const int*   __restrict__ tts,
                                                      const float* __restrict__ Mt,
                                                      const float* __restrict__ biasv,
                                                      float* __restrict__ out) {
    __shared__ float xs[2][16][CHUNK + XPAD];  // double-buffered 16-row x-chunk
    __shared__ float sc[16][N_SETS + 4];       // scores tile
    __shared__ float wgt[16][2];
    __shared__ int   widx[16][2];

    const int tid  = threadIdx.x;
    const int lane = tid & 31;
    const int w    = tid >> 5;             // n-tile 0..3
    const int l15  = lane & 15;
    const int half = lane >> 4;
    const int kb   = half * 2;
    const int n    = w * 16 + l15;
    const long tokbase = (long)blockIdx.x * 16;

    // staging map: thread -> (row, 32-float slab) ; 8 x b128 per thread per chunk
    const int lr = tid >> 3;               // row 0..15
    const int lc = (tid & 7) * 32;         // col base
    const char* gbase = (const char*)(x + (size_t)(tokbase + lr) * D_MODEL + lc);

    const float* mrow = Mt + (size_t)n * D_MODEL + kb;
    v8f acc = {};

#ifdef HAVE_ASYNC_LDS
#define ACP(GS, LS, OFF) \
    __builtin_amdgcn_global_load_async_to_lds_b128((AS1 i4*)((GS) + (OFF)), (AS3 i4*)((LS) + (OFF)), 0, 0)
#define ISSUE_CHUNK(CI, BUF)                                                          \
    do {                                                                              \
        const char* gs = gbase + (size_t)(CI) * (CHUNK * 4);                          \
        char* ls = (char*)&xs[(BUF)][lr][lc];                                         \
        ACP(gs, ls, 0);   ACP(gs, ls, 16);  ACP(gs, ls, 32);  ACP(gs, ls, 48);        \
        ACP(gs, ls, 64);  ACP(gs, ls, 80);  ACP(gs, ls, 96);  ACP(gs, ls, 112);       \
    } while (0)

    int buf = 0;
    ISSUE_CHUNK(0, 0);
    for (int ci = 0; ci < NCHUNK; ++ci) {
        if (ci + 1 < NCHUNK) {
            ISSUE_CHUNK(ci + 1, buf ^ 1);   // prefetch next chunk into other buffer
            WAIT_ASYNC(8);                  // chunk ci landed; next chunk in flight
        } else {
            WAIT_ASYNC(0);
        }
        __syncthreads();
        const int ch = ci * CHUNK;
#pragma unroll 8
        for (int kk = 0; kk < CHUNK; kk += 4) {
            v2f a = *(const v2f*)&xs[buf][l15][kk + kb];     // ds_load_b64
            v2f b = *(const v2f*)(mrow + ch + kk);           // global b64 (L2-hot)
            acc = __builtin_amdgcn_wmma_f32_16x16x4_f32(false, a, false, b,
                                                        (short)0, acc, false, false);
        }
        __syncthreads();                    // buffer free before it is re-filled
        buf ^= 1;
    }
#else
    // fallback: synchronous staging through VGPRs
    for (int ci = 0; ci < NCHUNK; ++ci) {
        const int ch = ci * CHUNK;
        __syncthreads();
        const float4* src = (const float4*)(gbase + (size_t)ch * 4);
#pragma unroll
        for (int j = 0; j < 8; ++j)
            *(float4*)&xs[0][lr][lc + j * 4] = src[j];
        __syncthreads();
#pragma unroll 8
        for (int kk = 0; kk < CHUNK; kk += 4) {
            v2f a = *(const v2f*)&xs[0][l15][kk + kb];
            v2f b = *(const v2f*)(mrow + ch + kk);
            acc = __builtin_amdgcn_wmma_f32_16x16x4_f32(false, a, false, b,
                                                        (short)0, acc, false, false);
        }
    }
#endif

    // scatter scores (+ router bias) to LDS: row = token-in-tile, col = set
    const float bv = biasv[n];
#pragma unroll
    for (int j = 0; j < 8; ++j)
        sc[j + 8 * half][n] = acc[j] + bv;
    __syncthreads();

    // per-token mask + top-2 + softmax (TOPK=2 over K=4 candidate sets)
    if (tid < 16) {
        const int sIdx = (int)((tokbase + tid) & (SEQ - 1));
        int   idx[4];
        float v[4];
#pragma unroll
        for (int j = 0; j < 4; ++j) {
            idx[j] = tts[sIdx * 4 + j];
            v[j]   = (idx[j] >= 0) ? sc[tid][idx[j]] : -3.402823e38f;
        }
        int jb = 0;
#pragma unroll
        for (int j = 1; j < 4; ++j) if (v[j] > v[jb]) jb = j;
        const int   n0 = idx[jb];
        const float v0 = v[jb];
        int n1 = n0; float v1 = -3.402823e38f;
#pragma unroll
        for (int j = 0; j < 4; ++j)
            if (idx[j] != n0 && v[j] > v1) { v1 = v[j]; n1 = idx[j]; }
        const float e1  = (n1 == n0) ? 0.f : __expf(v1 - v0);
        const float inv = 1.f / (1.f + e1);
        wgt[tid][0] = inv;  wgt[tid][1] = e1 * inv;
        widx[tid][0] = n0;  widx[tid][1] = (n1 == n0) ? n0 : n1;
    }
    __syncthreads();

    // out[b, s, :] = w0*set_states[b,i0,:] + w1*set_states[b,i1,:]
    const long bi = tokbase / SEQ;
    for (int t = 0; t < 16; ++t) {
        const float w0 = wgt[t][0], w1 = wgt[t][1];
        const float4* p0 = (const float4*)(set_states + ((size_t)bi * N_SETS + widx[t][0]) * D_MODEL);
        const float4* p1 = (const float4*)(set_states + ((size_t)bi * N_SETS + widx[t][1]) * D_MODEL);
        float4* po = (float4*)(out + (size_t)(tokbase + t) * D_MODEL);
        for (int c = tid; c < D_MODEL / 4; c += 128) {
            float4 a = p0[c], b = p1[c], r;
            r.x = w0 * a.x + w1 * b.x;
            r.y = w0 * a.y + w1 * b.y;
            r.z = w0 * a.z + w1 * b.z;
            r.w = w0 * a.w + w1 * b.w;
            po[c] = r;
        }
    }
}

extern "C" void kernel_launch(void* const* d_in, const int* in_sizes, int n_in,
                              void* d_out, int out_size, void* d_ws, size_t ws_size,
                              hipStream_t stream) {
    (void)in_sizes; (void)n_in; (void)out_size; (void)ws_size;
    const float* token_states = (const float*)d_in[0];  // [4,4096,2048]
    const float* set_states   = (const float*)d_in[1];  // [4,64,2048]
    const float* desc_router  = (const float*)d_in[2];  // [64,2048]
    const float* W_q          = (const float*)d_in[3];  // [2048,2048]
    const float* b_q          = (const float*)d_in[4];  // [2048]
    const int*   token_to_set = (const int*)d_in[5];    // [4096,4]
    float* out = (float*)d_out;

    float* Mt    = (float*)d_ws;                 // [64][2048] = 512 KB
    float* biasv = Mt + (size_t)N_SETS * D_MODEL;

    const float scale = 1.0f / sqrtf((float)D_MODEL);

    lr_precompute_M<<<dim3(128), dim3(128), 0, stream>>>(W_q, desc_router, Mt, scale);
    lr_bias<<<dim3(64), dim3(256), 0, stream>>>(b_q, desc_router, biasv, scale);
    lr_router_main<<<dim3(1024), dim3(128), 0, stream>>>(token_states, set_states,
                                                         token_to_set, Mt, biasv, out);
}